// KNNGroup_20486994002141
// MI455X (gfx1250) — compile-verified
//
#include <hip/hip_runtime.h>
#include <hip/hip_bf16.h>

typedef __attribute__((ext_vector_type(2))) float v2f;
typedef __attribute__((ext_vector_type(8))) float v8f;

#define BQ   16    // query rows per workgroup (one M-tile)
#define NC   256   // support columns per chunk (16 waves x 16 cols)
#define CST  18    // LDS column stride in floats (conflict-free: gcd(18,64)=2)
#define KNN  32    // neighbors == warpSize on gfx1250 (wave32)

__device__ __forceinline__ float readlane_f(float v, int l) {
  return __int_as_float(__builtin_amdgcn_readlane(__float_as_int(v), l));
}

__launch_bounds__(512)
__global__ void knn_group_kernel(const float* __restrict__ query_xyz,   // [B,M,3]
                                 const float* __restrict__ support_xyz, // [B,N,3]
                                 const float* __restrict__ features,    // [B,C,N]
                                 float* __restrict__ xyz_out,           // [B,3,M,K]
                                 float* __restrict__ feat_out,          // [B,C,M,K]
                                 int B_, int M_, int N_, int C_) {
  // Double-buffered LDS score tile, column-major: buf[col*CST + row]
  __shared__ float smem[2 * NC * CST];

  const int lane = threadIdx.x & 31;
  const int w    = threadIdx.x >> 5;          // wave id 0..15
  const int mtiles = M_ / BQ;
  const int b  = blockIdx.x / mtiles;
  const int m0 = (blockIdx.x % mtiles) * BQ;

  // ---- A operand (16x4 f32): row m = (qx, qy, qz, 1). Loop-invariant. ----
  // Layout: lanes 0-15 hold K=0 (VGPR0) / K=1 (VGPR1) for row=lane,
  //         lanes 16-31 hold K=2 / K=3 for row=lane-16.
  const int arow = lane & 15;
  const float* qp0 = query_xyz + ((size_t)b * M_ + (m0 + arow)) * 3;
  const float qx0 = qp0[0], qy0 = qp0[1], qz0 = qp0[2];
  v2f av;
  av.x = (lane < 16) ? qx0 : qz0;
  av.y = (lane < 16) ? qy0 : 1.0f;

  // ---- WMMA producer: score = |s|^2 - 2 q.s (same top-k order as d2) ----
  auto compute_tile = [&](int chunk, float* buf) {
    const int ncol = chunk * NC + w * 16 + (lane & 15);
    const float* sp = support_xyz + ((size_t)b * N_ + ncol) * 3;
    // speculative prefetch of the chunk after this one (global_prefetch_b8)
    __builtin_prefetch(sp + (size_t)NC * 3, 0, 1);
    const float sx = sp[0], sy = sp[1], sz = sp[2];
    // B operand (4x16 f32): col n = (-2sx, -2sy, -2sz, |s|^2)
    v2f bv;
    if (lane < 16) { bv.x = -2.0f * sx; bv.y = -2.0f * sy; }
    else           { bv.x = -2.0f * sz; bv.y = sx*sx + sy*sy + sz*sz; }

    v8f acc = {0.f, 0.f, 0.f, 0.f, 0.f, 0.f, 0.f, 0.f};
    acc = __builtin_amdgcn_wmma_f32_16x16x4_f32(false, av, false, bv,
                                                (short)0, acc, false, false);

    // D layout: VGPR i -> row i (lanes 0-15) / row i+8 (lanes 16-31), col = lane&15
    const int cc = w * 16 + (lane & 15);
    const int rb = (lane < 16) ? 0 : 8;
    v2f* dst = (v2f*)&buf[cc * CST + rb];     // 8B-aligned (cc*CST even, rb even)
#pragma unroll
    for (int j = 0; j < 4; ++j) {
      v2f t; t.x = acc[2 * j]; t.y = acc[2 * j + 1];
      dst[j] = t;
    }
  };

  // ---- running top-32 (sorted ascending across lanes: lane k = k-th best) ----
  float lv  = 3.0e38f;   // list value
  int   li  = 0;         // list index n
  float thr = 3.0e38f;   // worst kept (uniform, SGPR-resident via readlane)

  const int nchunks = N_ / NC;

  compute_tile(0, smem);          // prologue fill of buffer 0
  __syncthreads();

  for (int chunk = 0; chunk < nchunks; ++chunk) {
    // Producer for next chunk into the other buffer (overlaps with scan below)
    if (chunk + 1 < nchunks)
      compute_tile(chunk + 1, smem + ((chunk + 1) & 1) * (NC * CST));

    // ======== consumer: wave w scans query-row w over NC cols of buf[chunk&1] ====
    const float* cur = smem + (chunk & 1) * (NC * CST);
#pragma unroll
    for (int blk = 0; blk < NC / 32; ++blk) {
      const int cc2 = blk * 32 + lane;
      const float v = cur[cc2 * CST + w];      // stride-18 -> conflict-free

      unsigned mask = __builtin_amdgcn_ballot_w32(v < thr);
      while (mask) {                           // wave-uniform loop
        const int j = __builtin_ctz(mask);
        mask &= mask - 1;
        const float cv = readlane_f(v, j);     // v_readlane_b32 (uniform)
        if (cv < thr) {                        // uniform branch
          const int ci = chunk * NC + blk * 32 + j;
          // insertion position = #list elements strictly smaller
          const unsigned lt = __builtin_amdgcn_ballot_w32(lv < cv);
          const int pos = __builtin_popcount(lt);   // <= 31 since cv < list[31]
          const float pv = __shfl_up(lv, 1, 32);
          const int   pi = __shfl_up(li, 1, 32);
          if (lane == pos)     { lv = cv; li = ci; }
          else if (lane > pos) { lv = pv; li = pi; }
          thr = readlane_f(lv, 31);            // new worst kept
        }
      }
    }
    __syncthreads();   // single barrier per chunk (double-buffered)
  }

  // ======== output: wave w -> query m0+w, lane -> neighbor k ========
  const int m = m0 + w;
  const int idx = li;
  {
    const float* sp = support_xyz + ((size_t)b * N_ + idx) * 3;
    const float* qq = query_xyz   + ((size_t)b * M_ + m)   * 3;
    const size_t base = (((size_t)b * 3) * M_ + m) * KNN + lane;
    const size_t dstride = (size_t)M_ * KNN;
    xyz_out[base]               = sp[0] - qq[0];
    xyz_out[base + dstride]     = sp[1] - qq[1];
    xyz_out[base + 2 * dstride] = sp[2] - qq[2];
  }
  {
    const float* fsrc = features + (size_t)b * C_ * N_ + idx;
    float* fdst = feat_out + (((size_t)b * C_) * M_ + m) * KNN + lane;
    const size_t sstride = (size_t)N_;
    const size_t dstride = (size_t)M_ * KNN;
#pragma unroll 4
    for (int c = 0; c < C_; ++c) {
      fdst[(size_t)c * dstride] = fsrc[(size_t)c * sstride];  // coalesced store
    }
  }
}

extern "C" void kernel_launch(void* const* d_in, const int* in_sizes, int n_in,
                              void* d_out, int out_size, void* d_ws, size_t ws_size,
                              hipStream_t stream) {
  (void)in_sizes; (void)n_in; (void)d_ws; (void)ws_size; (void)out_size;
  const float* query_xyz   = (const float*)d_in[0];  // [B,M,3]
  const float* support_xyz = (const float*)d_in[1];  // [B,N,3]
  const float* features    = (const float*)d_in[2];  // [B,C,N]

  const int B = 4, M = 4096, N = 16384, C = 64;
  float* out      = (float*)d_out;
  float* xyz_out  = out;                                   // [B,3,M,K]
  float* feat_out = out + (size_t)B * 3 * M * KNN;         // [B,C,M,K]

  const int blocks = B * (M / BQ);   // 1024 workgroups, 16 waves each
  knn_group_kernel<<<blocks, 512, 0, stream>>>(query_xyz, support_xyz, features,
                                               xyz_out, feat_out, B, M, N, C);
}